// LearnedAttentionPooling_12919261626889
// MI455X (gfx1250) — compile-verified
//
#include <hip/hip_runtime.h>
#include <math.h>

// ---------------------------------------------------------------------------
// Problem constants (from the reference)
// ---------------------------------------------------------------------------
#define BB     8
#define MM     4096      // chunks (keys)
#define NN     4096      // points
#define DD     256
#define HH     4
#define DH     64
#define MOUT   196
#define NEGV   (-1e30f)

typedef __attribute__((ext_vector_type(2))) float v2f;
typedef __attribute__((ext_vector_type(4))) float v4f;
typedef __attribute__((ext_vector_type(8))) float v8f;

// ---------------------------------------------------------------------------
// Workspace layout (float offsets)
// ---------------------------------------------------------------------------
#define OFF_Q      0L                                   // 196*256
#define OFF_K      (OFF_Q + 196L*256L)                  // 8*4096*256
#define OFF_V      (OFF_K + 8L*4096L*256L)              // 8*4096*256
#define OFF_S      (OFF_V + 8L*4096L*256L)              // 8*4*196*4096
#define OFF_CTX    (OFF_S + 8L*4L*196L*4096L)           // 8*196*256
#define OFF_POOL   (OFF_CTX + 8L*196L*256L)             // 8*196*256
#define OFF_HID    (OFF_POOL + 8L*196L*256L)            // 8*196*256
#define OFF_CC     (OFF_HID + 8L*196L*256L)             // 8*4096*3
#define OFF_CS     (OFF_CC + 8L*4096L*3L)               // 8*4096

// ---------------------------------------------------------------------------
// 16x16 output tile over K via V_WMMA_F32_16X16X4_F32, K-step of 8.
//
// K-slot permutation trick: the WMMA reduces over its 4 K-slots, and the
// slot -> K mapping is arbitrary as long as A and B agree. We map
//   lanes 0-15  -> K values  k .. k+3
//   lanes 16-31 -> K values  k+4 .. k+7
// so ONE global_load_b128 per lane (A[m, k+4*(lane>>4) .. +3]) feeds TWO
// WMMAs: wmma0 consumes (.x,.y) = slots {k,k+1 | k+4,k+5}, wmma1 consumes
// (.z,.w) = slots {k+2,k+3 | k+6,k+7}. Same for B when it is given as
// W (N x K row-major). Out-of-range A rows are CLAMPED (not predicated),
// keeping EXEC all-1s and the inner loop branch-free; bad rows are simply
// never stored.
//
// BT=true : B given as W (N x K, row-major)  -> contiguous b128 loads
// BT=false: B given as K x N row-major       -> 4 strided scalar loads
// ---------------------------------------------------------------------------
template <bool BT>
__device__ inline v8f wave_tile_gemm(const float* __restrict__ A, int lda,
                                     int rowBase, int M,
                                     const float* __restrict__ B, int ldb,
                                     int colBase, int K, unsigned lane) {
    v8f c = {};
    int m = rowBase + (int)(lane & 15u);
    if (m > M - 1) m = M - 1;                 // clamp: keeps loads valid, EXEC full
    const int n    = colBase + (int)(lane & 15u);
    const int kgrp = (int)(lane >> 4u) * 4;   // 0 or 4
    const float* ap = A + (long)m * lda + kgrp;
    const float* bp = BT ? (B + (long)n * ldb + kgrp)
                         : (B + (long)kgrp * ldb + n);
    const long bstep = BT ? 8 : (long)8 * ldb;
    #pragma unroll 4
    for (int k = 0; k < K; k += 8) {
        const v4f a = *(const v4f*)ap;
        v2f a0, a1, b0, b1;
        a0.x = a.x; a0.y = a.y;
        a1.x = a.z; a1.y = a.w;
        if (BT) {
            const v4f b = *(const v4f*)bp;
            b0.x = b.x; b0.y = b.y;
            b1.x = b.z; b1.y = b.w;
        } else {
            b0.x = bp[0];
            b0.y = bp[(long)ldb];
            b1.x = bp[(long)2 * ldb];
            b1.y = bp[(long)3 * ldb];
        }
        c = __builtin_amdgcn_wmma_f32_16x16x4_f32(false, a0, false, b0,
                                                  (short)0, c, false, false);
        c = __builtin_amdgcn_wmma_f32_16x16x4_f32(false, a1, false, b1,
                                                  (short)0, c, false, false);
        ap += 8;
        bp += bstep;
    }
    return c;
}

// ---------------------------------------------------------------------------
// Generic batched GEMM: Out[z] = act( A[z] (MxK) * B[z] * scale + bias )
// grid = (ceil(N/64), ceil(M/32), Z); block = 256 (8 waves, 2x4 tile grid)
// z -> (b = z/hdiv, h = z%hdiv); per-operand (batch,head) strides.
// ---------------------------------------------------------------------------
template <bool BT, bool GELU>
__global__ void gemm_wmma(const float* __restrict__ A, int lda, long sAb, long sAh,
                          const float* __restrict__ Bm, int ldb, long sBb, long sBh,
                          const float* __restrict__ bias,
                          float* __restrict__ Out, int ldo, long sOb, long sOh,
                          int M, int N, int K, float scale, int hdiv) {
    const unsigned lane = threadIdx.x & 31u;
    const unsigned wave = threadIdx.x >> 5u;       // 0..7
    const int wm = (int)(wave >> 2u);              // 0..1
    const int wn = (int)(wave & 3u);               // 0..3
    const int rowBase = blockIdx.y * 32 + wm * 16;
    const int colBase = blockIdx.x * 64 + wn * 16;
    if (rowBase >= M || colBase >= N) return;      // wave-uniform exit

    const int z  = blockIdx.z;
    const int bb = z / hdiv;
    const int hh = z % hdiv;
    const float* Ap = A   + bb * sAb + hh * sAh;
    const float* Bp = Bm  + bb * sBb + hh * sBh;
    float*       Op = Out + bb * sOb + hh * sOh;

    v8f c = wave_tile_gemm<BT>(Ap, lda, rowBase, M, Bp, ldb, colBase, K, lane);

    const int n = colBase + (int)(lane & 15u);
    const float bv = bias ? bias[n] : 0.0f;
    const int rOff = (int)(lane >> 4u) * 8;
    #pragma unroll
    for (int i = 0; i < 8; ++i) {
        const int r = rowBase + rOff + i;
        if (r < M) {
            float v = c[i] * scale + bv;
            if (GELU) v = 0.5f * v * (1.0f + erff(v * 0.70710678118f));
            Op[(long)r * ldo + n] = v;
        }
    }
}

// ---------------------------------------------------------------------------
// Column reduce of assignments (B,N,M) against positions (B,N,3):
//   cs[b,m] = sum_n A[b,n,m];  cc[b,m,c] = (sum_n A[b,n,m]*P[b,n,c]) / clip(cs)
// One thread per column m, positions staged through LDS in 256-row chunks.
// assignments (512 MB) is streamed exactly once -> non-temporal loads so it
// does not evict the attention working set from the 192 MB L2.
// ---------------------------------------------------------------------------
__global__ void asg_reduce(const float* __restrict__ asg,
                           const float* __restrict__ pos,
                           float* __restrict__ cc, float* __restrict__ cs) {
    __shared__ float pl[256 * 3];
    const int b = blockIdx.y;
    const int m = blockIdx.x * blockDim.x + threadIdx.x;
    const float* Ab = asg + (long)b * NN * MM;
    const float* Pb = pos + (long)b * NN * 3;
    float sx = 0.f, sy = 0.f, sz = 0.f, ss = 0.f;
    for (int n0 = 0; n0 < NN; n0 += 256) {
        for (int t = threadIdx.x; t < 256 * 3; t += blockDim.x)
            pl[t] = Pb[(long)n0 * 3 + t];
        __syncthreads();
        #pragma unroll 4
        for (int j = 0; j < 256; ++j) {
            const float a = __builtin_nontemporal_load(&Ab[(long)(n0 + j) * MM + m]);
            ss += a;
            sx += a * pl[j * 3 + 0];
            sy += a * pl[j * 3 + 1];
            sz += a * pl[j * 3 + 2];
        }
        __syncthreads();
    }
    const float d = fmaxf(ss, 1e-6f);
    const long o = (long)b * MM + m;
    cc[o * 3 + 0] = sx / d;
    cc[o * 3 + 1] = sy / d;
    cc[o * 3 + 2] = sz / d;
    cs[o] = ss;
}

// ---------------------------------------------------------------------------
// Masked softmax over the key dim (4096), in place on S (B*H*Q rows).
// ---------------------------------------------------------------------------
__global__ void softmax_mask(float* __restrict__ S,
                             const unsigned char* __restrict__ mask) {
    __shared__ float red[256];
    const int row = blockIdx.x;                  // (b*H + h)*MOUT + q
    const int z   = row / MOUT;
    const int b   = z / HH;
    float* Sr = S + (long)row * MM;
    const unsigned char* mr = mask + (long)b * MM;
    const int tid = threadIdx.x;

    float lmax = -3.4e38f;
    for (int i = tid; i < MM; i += 256) {
        const float v = mr[i] ? NEGV : Sr[i];
        lmax = fmaxf(lmax, v);
    }
    red[tid] = lmax; __syncthreads();
    for (int s = 128; s > 0; s >>= 1) {
        if (tid < s) red[tid] = fmaxf(red[tid], red[tid + s]);
        __syncthreads();
    }
    const float bmax = red[0]; __syncthreads();

    float lsum = 0.f;
    for (int i = tid; i < MM; i += 256) {
        const float v = mr[i] ? NEGV : Sr[i];
        const float e = __expf(v - bmax);
        Sr[i] = e;
        lsum += e;
    }
    red[tid] = lsum; __syncthreads();
    for (int s = 128; s > 0; s >>= 1) {
        if (tid < s) red[tid] += red[tid + s];
        __syncthreads();
    }
    const float inv = 1.0f / red[0];
    for (int i = tid; i < MM; i += 256) Sr[i] *= inv;
}

// ---------------------------------------------------------------------------
// centroids[b,o,:] = sum_m mean_h(attn[b,h,o,m]) * cc[b,m,:]
// sizes[b,o]      = sum_m mean_h(attn[b,h,o,m]) * cs[b,m]
// One block per (b,o); 4-way block reduction.
// ---------------------------------------------------------------------------
__global__ void pool_geom(const float* __restrict__ S,
                          const float* __restrict__ cc,
                          const float* __restrict__ cs,
                          float* __restrict__ outC, float* __restrict__ outS) {
    __shared__ float red[4][256];
    const int idx = blockIdx.x;                  // b*MOUT + o
    const int b = idx / MOUT;
    const int o = idx % MOUT;
    const long hs = (long)MOUT * MM;
    const float* Sr = S + ((long)b * HH * MOUT + o) * MM;
    const float* ccb = cc + (long)b * MM * 3;
    const float* csb = cs + (long)b * MM;
    const int tid = threadIdx.x;

    float ax = 0.f, ay = 0.f, az = 0.f, as = 0.f;
    for (int m = tid; m < MM; m += 256) {
        const float w = 0.25f * (Sr[m] + Sr[m + hs] + Sr[m + 2 * hs] + Sr[m + 3 * hs]);
        ax += w * ccb[m * 3 + 0];
        ay += w * ccb[m * 3 + 1];
        az += w * ccb[m * 3 + 2];
        as += w * csb[m];
    }
    red[0][tid] = ax; red[1][tid] = ay; red[2][tid] = az; red[3][tid] = as;
    __syncthreads();
    for (int s = 128; s > 0; s >>= 1) {
        if (tid < s)
            for (int j = 0; j < 4; ++j) red[j][tid] += red[j][tid + s];
        __syncthreads();
    }
    if (tid == 0) {
        outC[(long)idx * 3 + 0] = red[0][0];
        outC[(long)idx * 3 + 1] = red[1][0];
        outC[(long)idx * 3 + 2] = red[2][0];
        outS[idx] = red[3][0];
    }
}

// ---------------------------------------------------------------------------
// Launch pipeline
// ---------------------------------------------------------------------------
extern "C" void kernel_launch(void* const* d_in, const int* in_sizes, int n_in,
                              void* d_out, int out_size, void* d_ws, size_t ws_size,
                              hipStream_t stream) {
    const float* z_chunks  = (const float*)d_in[0];
    const unsigned char* chunk_masks = (const unsigned char*)d_in[1];
    const float* positions = (const float*)d_in[2];
    const float* assigns   = (const float*)d_in[3];
    const float* query_tok = (const float*)d_in[4];
    const float* in_w      = (const float*)d_in[5];   // (768,256)
    const float* in_b      = (const float*)d_in[6];   // (768)
    const float* out_w     = (const float*)d_in[7];   // (256,256)
    const float* out_b     = (const float*)d_in[8];
    const float* w1        = (const float*)d_in[9];
    const float* b1        = (const float*)d_in[10];
    const float* w2        = (const float*)d_in[11];
    const float* b2        = (const float*)d_in[12];

    float* ws = (float*)d_ws;
    float* wsQ   = ws + OFF_Q;
    float* wsK   = ws + OFF_K;
    float* wsV   = ws + OFF_V;
    float* wsS   = ws + OFF_S;
    float* wsCtx = ws + OFF_CTX;
    float* wsPo  = ws + OFF_POOL;
    float* wsH   = ws + OFF_HID;
    float* wsCC  = ws + OFF_CC;
    float* wsCS  = ws + OFF_CS;

    float* outPooled = (float*)d_out;                       // (8,196,256)
    float* outCent   = outPooled + (long)BB * MOUT * DD;    // (8,196,3)
    float* outSize   = outCent + (long)BB * MOUT * 3;       // (8,196)

    const float* Wq = in_w;              const float* bq = in_b;
    const float* Wk = in_w + 256 * 256;  const float* bk = in_b + 256;
    const float* Wv = in_w + 512 * 256;  const float* bv = in_b + 512;

    // 1) geometry reduction over the 512 MB assignments tensor
    asg_reduce<<<dim3(MM / 256, BB, 1), 256, 0, stream>>>(assigns, positions, wsCC, wsCS);

    // 2) q projection (shared across batch): (196x256)@Wq^T + bq
    gemm_wmma<true, false><<<dim3(4, 7, 1), 256, 0, stream>>>(
        query_tok, DD, 0, 0, Wq, DD, 0, 0, bq,
        wsQ, DD, 0, 0, MOUT, DD, DD, 1.0f, 1);

    // 3) k / v projections: (32768x256)@W^T + b
    gemm_wmma<true, false><<<dim3(4, 1024, 1), 256, 0, stream>>>(
        z_chunks, DD, 0, 0, Wk, DD, 0, 0, bk,
        wsK, DD, 0, 0, BB * MM, DD, DD, 1.0f, 1);
    gemm_wmma<true, false><<<dim3(4, 1024, 1), 256, 0, stream>>>(
        z_chunks, DD, 0, 0, Wv, DD, 0, 0, bv,
        wsV, DD, 0, 0, BB * MM, DD, DD, 1.0f, 1);

    // 4) scores[b,h] = q_h (196x64) @ k_h^T (64x4096) / sqrt(64)
    gemm_wmma<true, false><<<dim3(64, 7, BB * HH), 256, 0, stream>>>(
        wsQ, DD, 0, (long)DH,
        wsK, DD, (long)MM * DD, (long)DH, nullptr,
        wsS, MM, (long)HH * MOUT * MM, (long)MOUT * MM,
        MOUT, MM, DH, 0.125f, HH);

    // 5) masked softmax over keys, in place
    softmax_mask<<<dim3(BB * HH * MOUT, 1, 1), 256, 0, stream>>>(wsS, chunk_masks);

    // 6) ctx[b,h] = attn (196x4096) @ v_h (4096x64)
    gemm_wmma<false, false><<<dim3(1, 7, BB * HH), 256, 0, stream>>>(
        wsS, MM, (long)HH * MOUT * MM, (long)MOUT * MM,
        wsV, DD, (long)MM * DD, (long)DH, nullptr,
        wsCtx, DD, (long)MOUT * DD, (long)DH,
        MOUT, DH, MM, 1.0f, HH);

    // 7) output head: pooled -> gelu(mlp1) -> mlp2  (1568x256x256 each)
    gemm_wmma<true, false><<<dim3(4, 49, 1), 256, 0, stream>>>(
        wsCtx, DD, 0, 0, out_w, DD, 0, 0, out_b,
        wsPo, DD, 0, 0, BB * MOUT, DD, DD, 1.0f, 1);
    gemm_wmma<true, true><<<dim3(4, 49, 1), 256, 0, stream>>>(
        wsPo, DD, 0, 0, w1, DD, 0, 0, b1,
        wsH, DD, 0, 0, BB * MOUT, DD, DD, 1.0f, 1);
    gemm_wmma<true, false><<<dim3(4, 49, 1), 256, 0, stream>>>(
        wsH, DD, 0, 0, w2, DD, 0, 0, b2,
        outPooled, DD, 0, 0, BB * MOUT, DD, DD, 1.0f, 1);

    // 8) head-averaged attention pooling of centroids & sizes
    pool_geom<<<dim3(BB * MOUT, 1, 1), 256, 0, stream>>>(wsS, wsCC, wsCS, outCent, outSize);
}